// RandWalkAggregator_23055384445492
// MI455X (gfx1250) — compile-verified
//
#include <hip/hip_runtime.h>

typedef float v2f __attribute__((ext_vector_type(2)));
typedef float v8f __attribute__((ext_vector_type(8)));

#define N_NODES 20000
#define KNB 32
#define DIM 128

__device__ __forceinline__ float fast_sigmoid(float x) {
    return 1.0f / (1.0f + __expf(-x));
}

// ---------------------------------------------------------------------------
// Kernel A: TSF = self_feats [N,128] @ self_weights [128,128]  (f32 WMMA)
// One block per 16-row tile; 8 waves; wave w owns output cols [16w, 16w+16).
// ---------------------------------------------------------------------------
__global__ __launch_bounds__(256)
void rw_self_gemm_kernel(const float* __restrict__ SF,
                         const float* __restrict__ SW,
                         float* __restrict__ TSF) {
    const int tid  = threadIdx.x;
    const int wave = tid >> 5;
    const int lane = tid & 31;
    const int h    = lane >> 4;   // lane half: selects K (A) / row (B) hi pair
    const int mn   = lane & 15;   // row (A-frag) or column (B/C/D-frag)
    const int rowBase = blockIdx.x * 16;
    const int colBase = wave * 16;

    v8f c = {};
    const float* arow = SF + (size_t)(rowBase + mn) * DIM;
    #pragma unroll 4
    for (int kk = 0; kk < DIM; kk += 4) {
        // A 16x4 frag: lane holds row mn, K = {kk+2h, kk+2h+1}
        v2f a = *(const v2f*)(arow + kk + 2 * h);
        // B 4x16 frag: VGPR0 row kk+2h, VGPR1 row kk+2h+1, col = colBase+mn
        v2f b;
        b.x = SW[(size_t)(kk + 2 * h)     * DIM + colBase + mn];
        b.y = SW[(size_t)(kk + 2 * h + 1) * DIM + colBase + mn];
        c = __builtin_amdgcn_wmma_f32_16x16x4_f32(false, a, false, b,
                                                  (short)0, c, false, false);
    }
    #pragma unroll
    for (int i = 0; i < 8; ++i) {
        TSF[(size_t)(rowBase + i + 8 * h) * DIM + colBase + mn] = c[i];
    }
}

// ---------------------------------------------------------------------------
// Kernel B: one block per node n.
//   gate[k]      = sigmoid(sv.gsw + nv[k].gnw + lv[k].glw)
//   T            = sigmoid(lv (32x128) @ LW (128x128))     [f32 WMMA]
//   out[e]       = relu( tsf[e] * (1/K) * sum_k nv[k,e]*T[k,e]*gate[k]/p[k] )
// ---------------------------------------------------------------------------
__global__ __launch_bounds__(256)
void rw_aggregate_kernel(const float* __restrict__ SV,
                         const float* __restrict__ NV,
                         const float* __restrict__ LV,
                         const float* __restrict__ P,
                         const float* __restrict__ GSW,
                         const float* __restrict__ GNW,
                         const float* __restrict__ GLW,
                         const float* __restrict__ LW,
                         const float* __restrict__ TSF,
                         float* __restrict__ OUT) {
    __shared__ float s_tnl[KNB];
    __shared__ float s_gd[KNB];
    __shared__ float s_acc[DIM];
    __shared__ float s_ts;

    const int n    = blockIdx.x;
    const int tid  = threadIdx.x;
    const int wave = tid >> 5;
    const int lane = tid & 31;
    const int h    = lane >> 4;
    const int mn   = lane & 15;

    const float* nv = NV + (size_t)n * KNB * DIM;
    const float* lv = LV + (size_t)n * KNB * DIM;

    if (tid < DIM) s_acc[tid] = 0.0f;

    // ---- gate dot products: wave w handles neighbors 4w..4w+3 -------------
    {
        const int d0 = lane * 4;              // 32 lanes x float4 = 128 elems
        const float4 gnw4 = *(const float4*)(GNW + d0);
        const float4 glw4 = *(const float4*)(GLW + d0);
        #pragma unroll
        for (int j = 0; j < 4; ++j) {
            const int k = wave * 4 + j;
            const float4 n4 = *(const float4*)(nv + (size_t)k * DIM + d0);
            const float4 l4 = *(const float4*)(lv + (size_t)k * DIM + d0);
            float s = n4.x * gnw4.x + n4.y * gnw4.y + n4.z * gnw4.z + n4.w * gnw4.w
                    + l4.x * glw4.x + l4.y * glw4.y + l4.z * glw4.z + l4.w * glw4.w;
            #pragma unroll
            for (int off = 16; off > 0; off >>= 1)
                s += __shfl_xor(s, off, 32);
            if (lane == 0) s_tnl[k] = s;
        }
        if (wave == 0) {
            const float4 sv4  = *(const float4*)(SV + (size_t)n * DIM + d0);
            const float4 gsw4 = *(const float4*)(GSW + d0);
            float s = sv4.x * gsw4.x + sv4.y * gsw4.y
                    + sv4.z * gsw4.z + sv4.w * gsw4.w;
            #pragma unroll
            for (int off = 16; off > 0; off >>= 1)
                s += __shfl_xor(s, off, 32);
            if (lane == 0) s_ts = s;
        }
    }
    __syncthreads();

    if (tid < KNB) {
        const float g = fast_sigmoid(s_ts + s_tnl[tid]);
        // fold importance weight and 1/K mean into one scale
        s_gd[tid] = g / (P[(size_t)n * KNB + tid] * (float)KNB);
    }
    __syncthreads();

    // ---- WMMA GEMM: T = lv (32x128) @ LW (128x128) -------------------------
    // wave owns columns [16*wave, 16*wave+16); two row tiles (rt=0,1) share
    // each B fragment.
    const int ct = wave;
    v8f c0 = {};
    v8f c1 = {};
    const float* a0row = lv + (size_t)mn * DIM;          // rows 0..15
    const float* a1row = lv + (size_t)(16 + mn) * DIM;   // rows 16..31
    #pragma unroll 4
    for (int kk = 0; kk < DIM; kk += 4) {
        v2f b;
        b.x = LW[(size_t)(kk + 2 * h)     * DIM + ct * 16 + mn];
        b.y = LW[(size_t)(kk + 2 * h + 1) * DIM + ct * 16 + mn];
        const v2f a0 = *(const v2f*)(a0row + kk + 2 * h);
        const v2f a1 = *(const v2f*)(a1row + kk + 2 * h);
        c0 = __builtin_amdgcn_wmma_f32_16x16x4_f32(false, a0, false, b,
                                                   (short)0, c0, false, false);
        c1 = __builtin_amdgcn_wmma_f32_16x16x4_f32(false, a1, false, b,
                                                   (short)0, c1, false, false);
    }

    // ---- fused epilogue: sigmoid, amplify, weighted mean over K -----------
    const int e = ct * 16 + mn;
    float psum = 0.0f;
    #pragma unroll
    for (int i = 0; i < 8; ++i) {
        const int k0 = i + 8 * h;        // rows of tile rt=0
        const int k1 = 16 + i + 8 * h;   // rows of tile rt=1
        psum += fast_sigmoid(c0[i]) * nv[(size_t)k0 * DIM + e] * s_gd[k0];
        psum += fast_sigmoid(c1[i]) * nv[(size_t)k1 * DIM + e] * s_gd[k1];
    }
    atomicAdd(&s_acc[e], psum);          // ds_add_f32: 2 adds per column
    __syncthreads();

    if (tid < DIM) {
        const float v = s_acc[tid] * TSF[(size_t)n * DIM + tid];
        OUT[(size_t)n * DIM + tid] = fmaxf(v, 0.0f);
    }
}

extern "C" void kernel_launch(void* const* d_in, const int* in_sizes, int n_in,
                              void* d_out, int out_size, void* d_ws, size_t ws_size,
                              hipStream_t stream) {
    const float* self_feats   = (const float*)d_in[0];
    const float* self_vecs    = (const float*)d_in[1];
    const float* neigh_vecs   = (const float*)d_in[2];
    const float* link_vecs    = (const float*)d_in[3];
    const float* select_probs = (const float*)d_in[4];
    const float* gate_self_w  = (const float*)d_in[5];
    const float* gate_neigh_w = (const float*)d_in[6];
    const float* gate_link_w  = (const float*)d_in[7];
    const float* self_weights = (const float*)d_in[8];
    const float* link_weights = (const float*)d_in[9];
    float* out = (float*)d_out;
    float* tsf = (float*)d_ws;   // N*D floats = 10.24 MB scratch

    rw_self_gemm_kernel<<<N_NODES / 16, 256, 0, stream>>>(self_feats, self_weights, tsf);
    rw_aggregate_kernel<<<N_NODES, 256, 0, stream>>>(self_vecs, neigh_vecs, link_vecs,
                                                     select_probs, gate_self_w,
                                                     gate_neigh_w, gate_link_w,
                                                     link_weights, tsf, out);
}